// VQVAE_80204219285886
// MI455X (gfx1250) — compile-verified
//
#include <hip/hip_runtime.h>

typedef __attribute__((ext_vector_type(2))) float v2f;
typedef __attribute__((ext_vector_type(8))) float v8f;

#define BATCH 8
#define CIN 3
#define CH 64
#define HT 64
#define WD 64
#define NSP (HT*WD)          // 4096 spatial
#define NPOS (BATCH*NSP)     // 32768 positions
#define KCB 4096             // codebook entries
#define NGRP 8
#define CPG (CH/NGRP)        // 8 channels per group

#define NTILE (KCB / 16)     // 256 column tiles
#define LDS_ROW 272          // 64 floats (256B) padded to 272B: banks = 4*col + off
#define LDS_TILE (16 * LDS_ROW)  // 4352 B per staged tile

#define ASYNC_WAIT(n) asm volatile("s_wait_asynccnt %0" :: "i"(n) : "memory")

// ---------------- Kernel 1: conv_in 3->64, output feat as [P, 64] ----------------
__global__ void conv_in_kernel(const float* __restrict__ x,
                               const float* __restrict__ w,
                               const float* __restrict__ bias,
                               float* __restrict__ feat) {
    __shared__ float sw[CH * 27];
    __shared__ float sb[CH];
    for (int i = threadIdx.x; i < CH * 27; i += blockDim.x) sw[i] = w[i];
    for (int i = threadIdx.x; i < CH; i += blockDim.x) sb[i] = bias[i];
    __syncthreads();
    int p = blockIdx.x * blockDim.x + threadIdx.x;
    int b = p >> 12;
    int hw = p & (NSP - 1);
    int h = hw >> 6, wc = hw & 63;
    float in[27];
#pragma unroll
    for (int ci = 0; ci < CIN; ++ci)
#pragma unroll
        for (int kh = 0; kh < 3; ++kh)
#pragma unroll
            for (int kw = 0; kw < 3; ++kw) {
                int hh = h + kh - 1, ww = wc + kw - 1;
                float v = 0.f;
                if (hh >= 0 && hh < HT && ww >= 0 && ww < WD)
                    v = x[((b * CIN + ci) * HT + hh) * WD + ww];
                in[ci * 9 + kh * 3 + kw] = v;
            }
    float* out = feat + (size_t)p * CH;
    for (int co = 0; co < CH; ++co) {
        float acc = sb[co];
        const float* wr = &sw[co * 27];
#pragma unroll
        for (int t = 0; t < 27; ++t) acc = fmaf(in[t], wr[t], acc);
        out[co] = acc;
    }
}

// ---------------- Kernel 1b: codebook squared norms ----------------
__global__ void cnorm_kernel(const float* __restrict__ cb, float* __restrict__ cnorm) {
    int k = blockIdx.x * blockDim.x + threadIdx.x;
    const float* r = cb + (size_t)k * CH;
    float s = 0.f;
#pragma unroll
    for (int c = 0; c < CH; ++c) s = fmaf(r[c], r[c], s);
    cnorm[k] = s;
}

// ------- Kernel 2: fp32 WMMA distance GEMM + argmin, async-LDS staged B -------
// Block = 256 threads = 8 waves; each wave owns 16 rows; block covers 128 rows.
// Codebook tiles (16 codes x 64ch = 4KB) staged to LDS via async copies, 2 ahead.
__global__ void __launch_bounds__(256) vq_argmin_kernel(
        const float* __restrict__ feat,
        const float* __restrict__ cb,
        const float* __restrict__ cnorm,
        int* __restrict__ min_idx) {
    __shared__ char lds[4 * LDS_TILE];   // 4-deep ring, padded rows

    int tid = threadIdx.x;
    int lane = tid & 31;
    int wave = tid >> 5;
    int col = lane & 15;
    int half = lane >> 4;  // 0: K={0,1}+4k  1: K={2,3}+4k
    int r0 = (blockIdx.x * 8 + wave) * 16;

    // Preload A operand (row-major feat) — 16 k-steps over 64 channels
    v2f a[16];
    const float* arow = feat + (size_t)(r0 + col) * CH + half * 2;
#pragma unroll
    for (int k = 0; k < 16; ++k) a[k] = *(const v2f*)(arow + 4 * k);

    float bestd[8];
    int besti[8];
#pragma unroll
    for (int j = 0; j < 8; ++j) { bestd[j] = 3.4e38f; besti[j] = 0; }

    // Async staging geometry: thread i moves one 16B chunk per tile.
    unsigned ldsBase = (unsigned)(size_t)(&lds[0]);         // low 32 bits = LDS offset
    int rowc = tid >> 4;                                    // codebook row in tile
    int chunk = tid & 15;                                   // 16B chunk in row
    unsigned dstOff = (unsigned)(rowc * LDS_ROW + chunk * 16);
    uint64_t srcBase = (uint64_t)(size_t)cb + (uint64_t)tid * 16;

    // Prologue: issue tiles 0 and 1
    {
        unsigned d0 = ldsBase + 0 * LDS_TILE + dstOff;
        uint64_t s0 = srcBase;
        asm volatile("global_load_async_to_lds_b128 %0, %1, off" :: "v"(d0), "v"(s0) : "memory");
        unsigned d1 = ldsBase + 1 * LDS_TILE + dstOff;
        uint64_t s1 = srcBase + 4096;
        asm volatile("global_load_async_to_lds_b128 %0, %1, off" :: "v"(d1), "v"(s1) : "memory");
    }

    auto consume = [&](int t) {
        const char* bt = &lds[(t & 3) * LDS_TILE];
        const float* brow = (const float*)(bt + col * LDS_ROW) + half * 2;
        v2f breg[16];
#pragma unroll
        for (int k = 0; k < 16; ++k) breg[k] = *(const v2f*)(brow + 4 * k);
        v8f acc = {};
#pragma unroll
        for (int k = 0; k < 16; ++k)
            acc = __builtin_amdgcn_wmma_f32_16x16x4_f32(
                false, a[k], false, breg[k], (short)0, acc, false, false);
        float cn = cnorm[t * 16 + col];
        int idx = t * 16 + col;
#pragma unroll
        for (int j = 0; j < 8; ++j) {
            float d = fmaf(-2.f, acc[j], cn);   // ||c||^2 - 2 x.c  (||x||^2 const per row)
            if (d < bestd[j]) { bestd[j] = d; besti[j] = idx; }
        }
    };

    for (int t = 0; t < NTILE - 2; ++t) {
        unsigned dn = ldsBase + ((t + 2) & 3) * LDS_TILE + dstOff;
        uint64_t sn = srcBase + (uint64_t)(t + 2) * 4096;
        asm volatile("global_load_async_to_lds_b128 %0, %1, off" :: "v"(dn), "v"(sn) : "memory");
        ASYNC_WAIT(2);      // in-order completion => tile t landed
        __syncthreads();    // publish all threads' chunks
        consume(t);
    }
    ASYNC_WAIT(1);
    __syncthreads();
    consume(NTILE - 2);
    ASYNC_WAIT(0);
    __syncthreads();
    consume(NTILE - 1);

    // Reduce best across the 16 lanes of each half (rows j + 8*half)
#pragma unroll
    for (int j = 0; j < 8; ++j) {
        float d = bestd[j];
        int idx = besti[j];
        for (int off = 8; off > 0; off >>= 1) {
            float od = __shfl_xor(d, off, 32);
            int oi = __shfl_xor(idx, off, 32);
            if (od < d || (od == d && oi < idx)) { d = od; idx = oi; }
        }
        bestd[j] = d;
        besti[j] = idx;
    }
    if (col == 0) {
        int mbase = r0 + half * 8;
#pragma unroll
        for (int j = 0; j < 8; ++j) min_idx[mbase + j] = besti[j];
    }
}

// ---- Kernel 3: gather codes -> quant (NCHW in d_out) + per-block MSE partials ----
__global__ void gather_loss_kernel(const float* __restrict__ feat,
                                   const float* __restrict__ cb,
                                   const int* __restrict__ min_idx,
                                   float* __restrict__ quant,
                                   float* __restrict__ partial) {
    __shared__ float red[256];
    int p = blockIdx.x * blockDim.x + threadIdx.x;
    int idx = min_idx[p];
    const float* qr = cb + (size_t)idx * CH;
    const float* fr = feat + (size_t)p * CH;
    int b = p >> 12;
    int hw = p & (NSP - 1);
    float* qo = quant + (size_t)b * (CH * NSP) + hw;
    float se = 0.f;
#pragma unroll
    for (int c = 0; c < CH; ++c) {
        float q = qr[c];
        float dv = q - fr[c];
        se = fmaf(dv, dv, se);
        qo[(size_t)c * NSP] = q;
    }
    red[threadIdx.x] = se;
    __syncthreads();
    for (int s = 128; s > 0; s >>= 1) {
        if (threadIdx.x < (unsigned)s) red[threadIdx.x] += red[threadIdx.x + s];
        __syncthreads();
    }
    if (threadIdx.x == 0) partial[blockIdx.x] = red[0];
}

// ---------------- Kernel 4: finalize scalar losses ----------------
__global__ void loss_finalize_kernel(const float* __restrict__ partial, int n,
                                     float* __restrict__ losses) {
    __shared__ float red[128];
    float s = 0.f;
    for (int i = threadIdx.x; i < n; i += blockDim.x) s += partial[i];
    red[threadIdx.x] = s;
    __syncthreads();
    for (int st = 64; st > 0; st >>= 1) {
        if (threadIdx.x < (unsigned)st) red[threadIdx.x] += red[threadIdx.x + st];
        __syncthreads();
    }
    if (threadIdx.x == 0) {
        float mse = red[0] / (float)((size_t)NPOS * CH);
        losses[0] = 1.0f * mse;   // commitment_loss (CODEBOOK_WEIGHT)
        losses[1] = 0.25f * mse;  // code_book_loss (BETA)
    }
}

// ---------------- Kernel 5: GroupNorm stats per (b, g) ----------------
__global__ void gn_stats_kernel(const float* __restrict__ quant,
                                float* __restrict__ stats) {
    int bg = blockIdx.x;
    int b = bg >> 3, g = bg & 7;
    const float* base = quant + (size_t)b * CH * NSP + (size_t)g * CPG * NSP;
    float s = 0.f, s2 = 0.f;
    for (int i = threadIdx.x; i < CPG * NSP; i += blockDim.x) {
        float v = base[i];
        s += v;
        s2 = fmaf(v, v, s2);
    }
    __shared__ float r1[256], r2[256];
    r1[threadIdx.x] = s;
    r2[threadIdx.x] = s2;
    __syncthreads();
    for (int st = 128; st > 0; st >>= 1) {
        if (threadIdx.x < (unsigned)st) {
            r1[threadIdx.x] += r1[threadIdx.x + st];
            r2[threadIdx.x] += r2[threadIdx.x + st];
        }
        __syncthreads();
    }
    if (threadIdx.x == 0) {
        float inv_n = 1.f / (float)(CPG * NSP);
        float mean = r1[0] * inv_n;
        float var = r2[0] * inv_n - mean * mean;
        stats[bg * 2] = mean;
        stats[bg * 2 + 1] = rsqrtf(var + 1e-5f);
    }
}

// ---------------- Kernel 6: normalize + SiLU ----------------
__global__ void gn_silu_kernel(const float* __restrict__ quant,
                               const float* __restrict__ stats,
                               const float* __restrict__ scale,
                               const float* __restrict__ bias,
                               float* __restrict__ act) {
    int i = blockIdx.x * blockDim.x + threadIdx.x;
    int c = (i >> 12) & 63;
    int b = i >> 18;
    int bg = b * NGRP + (c >> 3);
    float mean = stats[bg * 2], inv = stats[bg * 2 + 1];
    float v = fmaf((quant[i] - mean) * inv, scale[c], bias[c]);
    act[i] = v / (1.f + __expf(-v));
}

// ---------------- Kernel 7: conv_out 64->3 ----------------
__global__ void conv_out_kernel(const float* __restrict__ act,
                                const float* __restrict__ w,
                                const float* __restrict__ bias,
                                float* __restrict__ recon) {
    __shared__ float sw[3 * CH * 9];
    for (int i = threadIdx.x; i < 3 * CH * 9; i += blockDim.x) sw[i] = w[i];
    __syncthreads();
    int p = blockIdx.x * blockDim.x + threadIdx.x;
    int b = p >> 12;
    int hw = p & (NSP - 1);
    int h = hw >> 6, wc = hw & 63;
    float acc0 = bias[0], acc1 = bias[1], acc2 = bias[2];
    for (int ci = 0; ci < CH; ++ci) {
        const float* ap = act + (size_t)(b * CH + ci) * NSP;
#pragma unroll
        for (int kh = 0; kh < 3; ++kh) {
            int hh = h + kh - 1;
            if (hh < 0 || hh >= HT) continue;
#pragma unroll
            for (int kw = 0; kw < 3; ++kw) {
                int ww = wc + kw - 1;
                if (ww < 0 || ww >= WD) continue;
                float v = ap[hh * WD + ww];
                int wi = ci * 9 + kh * 3 + kw;
                acc0 = fmaf(v, sw[0 * 576 + wi], acc0);
                acc1 = fmaf(v, sw[1 * 576 + wi], acc1);
                acc2 = fmaf(v, sw[2 * 576 + wi], acc2);
            }
        }
    }
    recon[((size_t)b * 3 + 0) * NSP + hw] = acc0;
    recon[((size_t)b * 3 + 1) * NSP + hw] = acc1;
    recon[((size_t)b * 3 + 2) * NSP + hw] = acc2;
}

extern "C" void kernel_launch(void* const* d_in, const int* in_sizes, int n_in,
                              void* d_out, int out_size, void* d_ws, size_t ws_size,
                              hipStream_t stream) {
    (void)in_sizes; (void)n_in; (void)out_size; (void)ws_size;
    const float* x         = (const float*)d_in[0];
    const float* conv_in_w = (const float*)d_in[1];
    const float* conv_in_b = (const float*)d_in[2];
    const float* codebook  = (const float*)d_in[3];
    const float* gn_scale  = (const float*)d_in[4];
    const float* gn_bias   = (const float*)d_in[5];
    const float* dec_w     = (const float*)d_in[6];
    const float* dec_b     = (const float*)d_in[7];

    float* out    = (float*)d_out;
    float* recon  = out;                                        // [8,3,64,64]
    float* quant  = out + (size_t)BATCH * 3 * NSP;              // [8,64,64,64]
    float* losses = quant + (size_t)BATCH * CH * NSP;           // 2 scalars

    char* ws = (char*)d_ws;
    float* feat    = (float*)ws;                                           // P*64 f32
    int*   min_idx = (int*)(ws + sizeof(float) * (size_t)NPOS * CH);       // P i32
    float* act     = (float*)(ws + sizeof(float) * (size_t)NPOS * CH
                                 + sizeof(int) * (size_t)NPOS);            // P*64 f32
    float* partial = act + (size_t)NPOS * CH;                              // 128 f32
    float* stats   = partial + 256;                                        // 128 f32
    float* cnorm   = stats + 256;                                          // 4096 f32

    conv_in_kernel<<<NPOS / 256, 256, 0, stream>>>(x, conv_in_w, conv_in_b, feat);
    cnorm_kernel<<<KCB / 256, 256, 0, stream>>>(codebook, cnorm);
    // 8 waves/block * 16 rows/wave = 128 rows/block -> 256 blocks
    vq_argmin_kernel<<<NPOS / 128, 256, 0, stream>>>(feat, codebook, cnorm, min_idx);
    gather_loss_kernel<<<NPOS / 256, 256, 0, stream>>>(feat, codebook, min_idx, quant, partial);
    loss_finalize_kernel<<<1, 128, 0, stream>>>(partial, NPOS / 256, losses);
    gn_stats_kernel<<<BATCH * NGRP, 256, 0, stream>>>(quant, stats);
    gn_silu_kernel<<<(BATCH * CH * NSP) / 256, 256, 0, stream>>>(quant, stats, gn_scale, gn_bias, act);
    conv_out_kernel<<<NPOS / 256, 256, 0, stream>>>(act, dec_w, dec_b, recon);
}